// ClusterOverlap_12214886990028
// MI455X (gfx1250) — compile-verified
//
#include <hip/hip_runtime.h>
#include <math.h>

#define BN    8192   // points
#define ENC   256    // encoding dim
#define NC    25     // clusters
#define SN    8192   // samples
#define KK    25     // k (kth = 26th smallest, index 25)
#define EPSV  1e-5f

#define TILE_M 16          // samples per workgroup
#define CHUNK  64          // points (columns) per chunk
#define NCHUNK (BN / CHUNK)
#define KSTEPS (ENC / 4)   // f32 WMMA K=4
#define LDA    (ENC + 4)   // padded LDS row stride (floats) -> conflict-free b64 loads
#define CAP    64          // candidate buffer per sample (== CHUNK, cannot overflow)

typedef float v2f __attribute__((ext_vector_type(2)));
typedef float v8f __attribute__((ext_vector_type(8)));
typedef int   v4i __attribute__((vector_size(16)));   // matches builtin prototype

#define GLOBAL_AS __attribute__((address_space(1)))
#define LDS_AS    __attribute__((address_space(3)))

#if __has_builtin(__builtin_amdgcn_global_load_async_to_lds_b128) && \
    __has_builtin(__builtin_amdgcn_s_wait_asynccnt)
#define USE_ASYNC_LDS 1
#else
#define USE_ASYNC_LDS 0
#endif

// ---------------- Phase 0: labels, confidence, squared norms ----------------
__global__ __launch_bounds__(256) void prep_kernel(
    const float* __restrict__ enc, const float* __restrict__ cat,
    int* __restrict__ labels, float* __restrict__ conf, float* __restrict__ e2) {
  int b = blockIdx.x * blockDim.x + threadIdx.x;
  if (b >= BN) return;
  const float* c = cat + (size_t)b * NC;
  float mx = c[0]; int am = 0;
  #pragma unroll
  for (int j = 1; j < NC; ++j) { float v = c[j]; if (v > mx) { mx = v; am = j; } }
  labels[b] = am;
  conf[b]   = mx;
  const float4* e = (const float4*)(enc + (size_t)b * ENC);
  float s = 0.f;
  #pragma unroll 8
  for (int j = 0; j < ENC / 4; ++j) {
    float4 v = e[j];
    s += v.x * v.x + v.y * v.y + v.z * v.z + v.w * v.w;
  }
  e2[b] = s;
}

// ---------------- Phase 1: fused GEMM + streaming top-26 + entropy ----------------
__global__ __launch_bounds__(128) void knn_entropy_kernel(
    const float* __restrict__ enc, const int* __restrict__ idxs,
    const int* __restrict__ labels, const float* __restrict__ conf,
    const float* __restrict__ e2, float* __restrict__ out) {
  __shared__ float    As[TILE_M * LDA];        // 16 query rows
  __shared__ float    Bs[2][CHUNK * LDA];      // double-buffered point chunk
  __shared__ float    candV[TILE_M][CAP];
  __shared__ int      candI[TILE_M][CAP];
  __shared__ unsigned ccnt[TILE_M];
  __shared__ float    Ts[TILE_M];              // running 26th-smallest threshold
  __shared__ float    q2s[TILE_M];
  __shared__ int      sIdx[TILE_M];

  const int tid  = threadIdx.x;
  const int w    = tid >> 5;        // wave id (0..3) -> 16-col subtile
  const int lane = tid & 31;        // wave32
  const int half = lane >> 4;       // 0: lanes 0-15, 1: lanes 16-31
  const int lr   = lane & 15;

  const int s0 = blockIdx.x * TILE_M;

  // private top-26 list (only threads 0..15 use it)
  float Lv[KK + 1];
  int   Li[KK + 1];
  #pragma unroll
  for (int i = 0; i <= KK; ++i) { Lv[i] = INFINITY; Li[i] = 0; }

  if (tid < TILE_M) {
    int gi = idxs[s0 + tid];
    sIdx[tid] = gi;
    q2s[tid]  = e2[gi];
    Ts[tid]   = INFINITY;
    ccnt[tid] = 0u;
  }
  __syncthreads();

  // ---- stage a 64-row B chunk into LDS buffer `buf` ----
  // Each of 128 threads moves 32 float4s: 64 rows x 64 float4/row.
  auto stage_chunk = [&](int ch, int buf) {
    const float* src = enc + (size_t)ch * CHUNK * ENC;
    #pragma unroll
    for (int j = 0; j < (CHUNK * (ENC / 4)) / 128; ++j) {
      int i4 = tid + j * 128;
      int r  = i4 >> 6;          // 64 float4 per row
      int c4 = i4 & 63;
      const float* g = src + (size_t)r * ENC + c4 * 4;
      float*       l = &Bs[buf][r * LDA + c4 * 4];
#if USE_ASYNC_LDS
      // CDNA5 async copy: memory -> LDS without a VGPR round trip (ASYNCcnt)
      __builtin_amdgcn_global_load_async_to_lds_b128(
          (GLOBAL_AS v4i*)(g), (LDS_AS v4i*)(l), 0, 0);
#else
      *(float4*)l = *(const float4*)g;
#endif
    }
  };

  // gather 16 query rows into LDS (float4, coalesced per row)
  for (int i4 = tid; i4 < TILE_M * (ENC / 4); i4 += 128) {
    int m  = i4 >> 6;
    int c4 = i4 & 63;
    float4 v = *(const float4*)(enc + (size_t)sIdx[m] * ENC + c4 * 4);
    *(float4*)(&As[m * LDA + c4 * 4]) = v;
  }

  // prime the pipeline with chunk 0
  stage_chunk(0, 0);

  for (int ch = 0; ch < NCHUNK; ++ch) {
    const int buf     = ch & 1;
    const int colBase = ch * CHUNK;

    // prefetch next chunk into the other buffer while we compute this one
    if (ch + 1 < NCHUNK) {
      stage_chunk(ch + 1, buf ^ 1);
#if USE_ASYNC_LDS
      // per-wave in-order completion: <=32 outstanding means chunk `ch` landed
      __builtin_amdgcn_s_wait_asynccnt(32);
#endif
    } else {
#if USE_ASYNC_LDS
      __builtin_amdgcn_s_wait_asynccnt(0);
#endif
    }
    __syncthreads();   // chunk `ch` visible to all waves

    // GEMM: each wave computes a 16x16 f32 tile via V_WMMA_F32_16X16X4_F32
    // A 16x4: lanes 0-15 hold {K0,K1} of row lr, lanes 16-31 hold {K2,K3}
    // B 4x16: same striping with column lr
    v8f acc = {0.f, 0.f, 0.f, 0.f, 0.f, 0.f, 0.f, 0.f};
    const float* Ap = &As[lr * LDA + 2 * half];
    const float* Bp = &Bs[buf][(w * 16 + lr) * LDA + 2 * half];
    #pragma unroll 8
    for (int k = 0; k < KSTEPS; ++k) {
      v2f a = *(const v2f*)(Ap + 4 * k);
      v2f b = *(const v2f*)(Bp + 4 * k);
      acc = __builtin_amdgcn_wmma_f32_16x16x4_f32(
          false, a, false, b, (short)0, acc, false, false);
    }

    // d2 = q2 + e2 - 2*dot ; filter against running threshold, append candidates
    const int   col = colBase + w * 16 + lr;
    const float e2n = e2[col];
    #pragma unroll
    for (int v = 0; v < 8; ++v) {
      int   m  = v + 8 * half;      // C layout: VGPR v -> row v (+8 for upper half)
      float d2 = q2s[m] + e2n - 2.0f * acc[v];
      d2 = fmaxf(d2, 0.0f);
      if (d2 < Ts[m]) {             // strict: boundary ties never needed
        unsigned pos = atomicAdd(&ccnt[m], 1u);
        if (pos < CAP) { candV[m][pos] = d2; candI[m][pos] = col; }
      }
    }
    __syncthreads();

    // merge candidates into private sorted top-26, tighten threshold.
    // (this barrier pair also fences re-use of buffer `buf` two chunks ahead)
    if (tid < TILE_M) {
      int c = (int)ccnt[tid]; if (c > CAP) c = CAP;
      for (int j = 0; j < c; ++j) {
        float vv = candV[tid][j];
        if (vv < Lv[KK]) {
          int ii = candI[tid][j];
          int p = KK;
          while (p > 0 && Lv[p - 1] > vv) { Lv[p] = Lv[p - 1]; Li[p] = Li[p - 1]; --p; }
          Lv[p] = vv; Li[p] = ii;
        }
      }
      Ts[tid]   = Lv[KK];
      ccnt[tid] = 0u;
    }
    __syncthreads();
  }

  // finalize: kth = 26th smallest; strict-less mask -> label histogram -> entropy
  if (tid < TILE_M) {
    const float kth = Lv[KK];
    int hist[NC];
    #pragma unroll
    for (int c = 0; c < NC; ++c) hist[c] = 0;
    int n = 0;
    for (int i = 0; i <= KK; ++i) {
      if (Lv[i] < kth) { ++hist[labels[Li[i]]]; ++n; }
    }
    float purity = 0.f;
    if (n > 0) {
      float inv = 1.0f / (float)n;
      for (int c = 0; c < NC; ++c) {
        float p = (float)hist[c] * inv;
        purity -= p * logf(p + EPSV);
      }
    }
    out[s0 + tid] = purity * conf[sIdx[tid]];
  }
}

extern "C" void kernel_launch(void* const* d_in, const int* in_sizes, int n_in,
                              void* d_out, int out_size, void* d_ws, size_t ws_size,
                              hipStream_t stream) {
  const float* enc  = (const float*)d_in[0];   // [B, ENC] f32
  const float* cat  = (const float*)d_in[1];   // [B, C]   f32
  const int*   idxs = (const int*)d_in[2];     // [S]      i32
  float*       out  = (float*)d_out;           // [S]      f32

  char* ws = (char*)d_ws;
  int*   labels = (int*)ws;                                    // B ints
  float* conf   = (float*)(ws + (size_t)BN * sizeof(int));     // B floats
  float* e2     = (float*)(ws + (size_t)BN * (sizeof(int) + sizeof(float)));

  prep_kernel<<<BN / 256, 256, 0, stream>>>(enc, cat, labels, conf, e2);
  knn_entropy_kernel<<<SN / TILE_M, 128, 0, stream>>>(enc, idxs, labels, conf, e2, out);
}